// PDA_GNN_6313601925678
// MI455X (gfx1250) — compile-verified
//
#include <hip/hip_runtime.h>
#include <hip/hip_bf16.h>

typedef float v2f __attribute__((ext_vector_type(2)));
typedef float v8f __attribute__((ext_vector_type(8)));

#define D 64

// ---------------- utility kernels ----------------

__global__ void pda_zero_kernel(float* __restrict__ p, size_t n) {
    size_t i = (size_t)blockIdx.x * blockDim.x + threadIdx.x;
    size_t stride = (size_t)gridDim.x * blockDim.x;
    for (; i < n; i += stride) p[i] = 0.0f;
}

__global__ void pda_deg_kernel(const int* __restrict__ dst, float* __restrict__ deg, int E) {
    int e = blockIdx.x * blockDim.x + threadIdx.x;
    if (e < E) atomicAdd(deg + dst[e], 1.0f);
}

__global__ void pda_dinv_kernel(float* __restrict__ deg, int N) {
    int i = blockIdx.x * blockDim.x + threadIdx.x;
    if (i < N) {
        float d = deg[i];
        deg[i] = (d > 0.0f) ? (1.0f / sqrtf(d)) : 0.0f;
    }
}

// One edge handled by 16 consecutive threads, each moving a float4 chunk of
// the 64-wide feature row: coalesced 256B gather of x[src], atomic scatter-add
// into out[dst]. norm = dinv[src]*dinv[dst] (symmetric gcn_norm, no self loops).
__global__ void pda_lgconv_kernel(const float* __restrict__ x,
                                  const int* __restrict__ src,
                                  const int* __restrict__ dst,
                                  const float* __restrict__ dinv,
                                  float* __restrict__ out, int E) {
    long g = (long)blockIdx.x * blockDim.x + threadIdx.x;
    long e = g >> 4;
    int c = (int)(g & 15);
    if (e >= E) return;
    int s = src[e];
    int d = dst[e];
    float nrm = dinv[s] * dinv[d];
    const float4 v = ((const float4*)(x + (long)s * D))[c];
    float* o = out + (long)d * D + c * 4;
    atomicAdd(o + 0, nrm * v.x);
    atomicAdd(o + 1, nrm * v.y);
    atomicAdd(o + 2, nrm * v.z);
    atomicAdd(o + 3, nrm * v.w);
}

// One wave32 per node: 2 feats/lane, shfl reduction, x /= max(||x||, 1e-12).
__global__ void pda_l2norm_kernel(float* __restrict__ x, int N) {
    int node = blockIdx.x * 8 + (threadIdx.x >> 5);
    int lane = threadIdx.x & 31;
    if (node >= N) return;
    float* row = x + (long)node * D;
    float a = row[lane];
    float b = row[lane + 32];
    float s = a * a + b * b;
    #pragma unroll
    for (int off = 16; off; off >>= 1) s += __shfl_xor(s, off, 32);
    float scale = 1.0f / fmaxf(sqrtf(s), 1e-12f);
    row[lane] = a * scale;
    row[lane + 32] = b * scale;
}

// ---------------- fused attention MLP (WMMA f32 16x16x4) ----------------
// One wave computes a 16-node tile. h[16][128] is held as 8 v8f accumulators,
// built with V_WMMA_F32_16X16X4_F32 over K = 192 (title|year|cat concat).
// A-frag (16x4 f32): lanes 0-15 carry K=k,k+1 for M=lane; lanes 16-31 carry
// K=k+2,k+3 for M=lane-16. B-frag (4x16): VGPR0 = rows k (lanes 0-15) /
// k+2 (lanes 16-31), VGPR1 = rows k+1 / k+3, column = nt*16 + (lane&15).
// D layout: acc[nt][r] @ lane L == h[r + 8*(L>=16)][nt*16 + (L&15)].
__global__ void pda_attn_mlp_kernel(const float* __restrict__ xt,
                                    const float* __restrict__ xy,
                                    const float* __restrict__ xc,
                                    const float* __restrict__ W1,  // [192][128]
                                    const float* __restrict__ b1,  // [128]
                                    const float* __restrict__ W2,  // [128][3]
                                    const float* __restrict__ b2,  // [3]
                                    float* __restrict__ out, int N) {
    const int lane = threadIdx.x & 31;
    const int wave = threadIdx.x >> 5;
    const int tile = blockIdx.x * 8 + wave;
    const int base = tile * 16;
    if (base >= N) return;  // wave-uniform exit; surviving waves keep EXEC all-1s
    const int half = lane >> 4;
    const int l16 = lane & 15;

    int arow = base + l16;
    if (arow >= N) arow = N - 1;  // clamped (safe) loads; stores guarded later
    const float* aptr[3] = { xt + (long)arow * D,
                             xy + (long)arow * D,
                             xc + (long)arow * D };

    v8f acc[8];
    #pragma unroll
    for (int nt = 0; nt < 8; nt++) {
        v8f z = { 0.f, 0.f, 0.f, 0.f, 0.f, 0.f, 0.f, 0.f };
        acc[nt] = z;
    }

    #pragma unroll
    for (int kb = 0; kb < 3; kb++) {
        const float* ap = aptr[kb];
        for (int k = 0; k < D; k += 4) {
            const int kk = k + 2 * half;
            v2f afrag;
            afrag[0] = ap[kk];
            afrag[1] = ap[kk + 1];
            const int krow = kb * D + kk;  // B rows krow, krow+1 for this lane
            #pragma unroll
            for (int nt = 0; nt < 8; nt++) {
                v2f bfrag;
                bfrag[0] = W1[(long)krow * 128 + nt * 16 + l16];
                bfrag[1] = W1[(long)(krow + 1) * 128 + nt * 16 + l16];
                acc[nt] = __builtin_amdgcn_wmma_f32_16x16x4_f32(
                    false, afrag, false, bfrag, (short)0, acc[nt], false, false);
            }
        }
    }

    // bias + relu (column c = nt*16 + l16 for every accumulator row)
    #pragma unroll
    for (int nt = 0; nt < 8; nt++) {
        float bc = b1[nt * 16 + l16];
        #pragma unroll
        for (int r = 0; r < 8; r++)
            acc[nt][r] = fmaxf(acc[nt][r] + bc, 0.0f);
    }

    // per-lane W2 columns for this lane's held h-columns
    float w2c[8][3];
    #pragma unroll
    for (int nt = 0; nt < 8; nt++) {
        int c = nt * 16 + l16;
        w2c[nt][0] = W2[c * 3 + 0];
        w2c[nt][1] = W2[c * 3 + 1];
        w2c[nt][2] = W2[c * 3 + 2];
    }
    const float bb0 = b2[0], bb1 = b2[1], bb2 = b2[2];

    for (int r = 0; r < 8; r++) {
        float p0 = 0.f, p1 = 0.f, p2 = 0.f;
        #pragma unroll
        for (int nt = 0; nt < 8; nt++) {
            float h = acc[nt][r];
            p0 += h * w2c[nt][0];
            p1 += h * w2c[nt][1];
            p2 += h * w2c[nt][2];
        }
        // reduce across the 16-lane half: together they cover all 128 columns
        #pragma unroll
        for (int off = 8; off; off >>= 1) {
            p0 += __shfl_xor(p0, off, 16);
            p1 += __shfl_xor(p1, off, 16);
            p2 += __shfl_xor(p2, off, 16);
        }
        float z0 = p0 + bb0, z1 = p1 + bb1, z2 = p2 + bb2;
        float mz = fmaxf(z0, fmaxf(z1, z2));
        float e0 = expf(z0 - mz), e1 = expf(z1 - mz), e2 = expf(z2 - mz);
        float inv = 1.0f / (e0 + e1 + e2);
        float a0 = e0 * inv, a1 = e1 * inv, a2 = e2 * inv;

        int node = base + r + 8 * half;
        if (node < N) {
            const float* t = xt + (long)node * D;
            const float* y = xy + (long)node * D;
            const float* c = xc + (long)node * D;
            float* o = out + (long)node * D;
            #pragma unroll
            for (int ff = 0; ff < 4; ff++) {
                int f = l16 + 16 * ff;
                o[f] = a0 * t[f] + a1 * y[f] + a2 * c[f];
            }
        }
    }
}

// ---------------- host ----------------

extern "C" void kernel_launch(void* const* d_in, const int* in_sizes, int n_in,
                              void* d_out, int out_size, void* d_ws, size_t ws_size,
                              hipStream_t stream) {
    const float* init = (const float*)d_in[0];
    const float* W1   = (const float*)d_in[1];
    const float* b1   = (const float*)d_in[2];
    const float* W2   = (const float*)d_in[3];
    const float* b2   = (const float*)d_in[4];
    const int* ei_t   = (const int*)d_in[5];
    const int* ei_y   = (const int*)d_in[6];
    const int* ei_c   = (const int*)d_in[7];
    float* out = (float*)d_out;

    const int N = in_sizes[0] / D;
    const int E = in_sizes[5] / 2;

    float* deg = (float*)d_ws;          // [N]      dinv scratch
    float* A   = deg + N;               // [N*64]   temp / x_year
    float* B   = A + (size_t)N * D;     // [N*64]   x_title
    float* C1  = B + (size_t)N * D;     // [N*64]   x_cat
    float* C2  = C1 + (size_t)N * D;    // [N*64]   cat temp

    const int TB = 256;
    const int zgrid = 2048;
    const int egrid = (E + TB - 1) / TB;
    const int ngrid = (N + TB - 1) / TB;
    const long cthreads = (long)E * 16;
    const int cgrid = (int)((cthreads + TB - 1) / TB);
    const int l2grid = (N + 7) / 8;
    const int tiles = (N + 15) / 16;
    const int mlpgrid = (tiles + 7) / 8;
    const size_t ND = (size_t)N * D;

    // ---- title branch: 2 convs ----
    {
        const int* src = ei_t;
        const int* dst = ei_t + E;
        pda_zero_kernel<<<zgrid, TB, 0, stream>>>(deg, (size_t)N);
        pda_deg_kernel<<<egrid, TB, 0, stream>>>(dst, deg, E);
        pda_dinv_kernel<<<ngrid, TB, 0, stream>>>(deg, N);
        pda_zero_kernel<<<zgrid, TB, 0, stream>>>(A, ND);
        pda_lgconv_kernel<<<cgrid, TB, 0, stream>>>(init, src, dst, deg, A, E);
        pda_zero_kernel<<<zgrid, TB, 0, stream>>>(B, ND);
        pda_lgconv_kernel<<<cgrid, TB, 0, stream>>>(A, src, dst, deg, B, E);
        pda_l2norm_kernel<<<l2grid, TB, 0, stream>>>(B, N);
    }
    // ---- year branch: 1 conv (reuse A) ----
    {
        const int* src = ei_y;
        const int* dst = ei_y + E;
        pda_zero_kernel<<<zgrid, TB, 0, stream>>>(deg, (size_t)N);
        pda_deg_kernel<<<egrid, TB, 0, stream>>>(dst, deg, E);
        pda_dinv_kernel<<<ngrid, TB, 0, stream>>>(deg, N);
        pda_zero_kernel<<<zgrid, TB, 0, stream>>>(A, ND);
        pda_lgconv_kernel<<<cgrid, TB, 0, stream>>>(init, src, dst, deg, A, E);
        pda_l2norm_kernel<<<l2grid, TB, 0, stream>>>(A, N);
    }
    // ---- cat branch: 3 convs ----
    {
        const int* src = ei_c;
        const int* dst = ei_c + E;
        pda_zero_kernel<<<zgrid, TB, 0, stream>>>(deg, (size_t)N);
        pda_deg_kernel<<<egrid, TB, 0, stream>>>(dst, deg, E);
        pda_dinv_kernel<<<ngrid, TB, 0, stream>>>(deg, N);
        pda_zero_kernel<<<zgrid, TB, 0, stream>>>(C1, ND);
        pda_lgconv_kernel<<<cgrid, TB, 0, stream>>>(init, src, dst, deg, C1, E);
        pda_zero_kernel<<<zgrid, TB, 0, stream>>>(C2, ND);
        pda_lgconv_kernel<<<cgrid, TB, 0, stream>>>(C1, src, dst, deg, C2, E);
        pda_zero_kernel<<<zgrid, TB, 0, stream>>>(C1, ND);
        pda_lgconv_kernel<<<cgrid, TB, 0, stream>>>(C2, src, dst, deg, C1, E);
        pda_l2norm_kernel<<<l2grid, TB, 0, stream>>>(C1, N);
    }
    // ---- fused attention MLP + blend (WMMA) ----
    pda_attn_mlp_kernel<<<mlpgrid, TB, 0, stream>>>(B, A, C1, W1, b1, W2, b2, out, N);
}